// DetectNet_44203803410566
// MI455X (gfx1250) — compile-verified
//
#include <hip/hip_runtime.h>
#include <hip/hip_bf16.h>
#include <stdint.h>

typedef __attribute__((ext_vector_type(2))) float v2f;
typedef __attribute__((ext_vector_type(8))) float v8f;

// ---------------- shapes ----------------
#define C1_OH 196
#define C1_OW 296
#define C1_PIX (C1_OH * C1_OW)     // 58016
#define C1_TILES (C1_PIX / 16)     // 3626
#define C1_K 300                   // 10*10*3

#define P1_H 98
#define P1_W 148

#define C2_OH 32
#define C2_OW 49

#define P2_H 16
#define P2_W 24

#define C3_OH 5
#define C3_OW 8

#define DK 5120
#define DCOLS 38400
#define DCOLS4 (DCOLS / 4)         // 9600
#define SPLITK 16
#define KCH (DK / SPLITK)          // 320

#define NB 4800                    // N_BOXES
#define VL 8                       // VEC_LEN
#define SORTN 8192

// async global->LDS path (gfx1250), guarded so a missing builtin degrades
// gracefully to a plain LDS store instead of failing the build.
#if defined(__has_builtin)
#if __has_builtin(__builtin_amdgcn_global_load_async_to_lds_b32) && \
    __has_builtin(__builtin_amdgcn_s_wait_asynccnt)
#define HAVE_ASYNC_LDS 1
#endif
#endif

typedef __attribute__((address_space(1))) int gbl_int;
typedef __attribute__((address_space(3))) int lds_int;

// =====================================================================
// conv1 as im2col GEMM with V_WMMA_F32_16X16X4_F32.
// A: 16 output pixels x 4 K-values, B: 4 K-values x 16 out-channels.
// Each wave: one M-tile, two N-halves (32 channels total).
// =====================================================================
__global__ __launch_bounds__(256) void conv1_wmma(
    const float* __restrict__ img, const float* __restrict__ w1,
    const float* __restrict__ b1, float* __restrict__ out) {
  __shared__ int offs[C1_K + 4];
  for (int t = threadIdx.x; t < C1_K; t += 256) {
    int kh = t / 30, r = t % 30, kw = r / 3, c = r % 3;
    offs[t] = kh * 1800 + kw * 3 + c;  // (kh*600 + kw)*3 + c
  }
  __syncthreads();

  const int wave = threadIdx.x >> 5;
  const int lane = threadIdx.x & 31;
  const int tile = blockIdx.x * 8 + wave;
  if (tile >= C1_TILES) return;  // uniform per wave

  const int m  = lane & 15;
  const int hi = lane >> 4;
  const int p  = tile * 16 + m;
  const int oy = p / C1_OW, ox = p % C1_OW;
  const float* ibase = img + (size_t)(oy * 2 * 600 + ox * 2) * 3;

  v8f acc0 = {};
  v8f acc1 = {};
  for (int kb = 0; kb < C1_K; kb += 4) {
    const int k0 = kb + hi * 2;
    v2f a, b0, b1v;
    a.x   = ibase[offs[k0]];
    a.y   = ibase[offs[k0 + 1]];
    b0.x  = w1[k0 * 32 + m];
    b0.y  = w1[(k0 + 1) * 32 + m];
    b1v.x = w1[k0 * 32 + 16 + m];
    b1v.y = w1[(k0 + 1) * 32 + 16 + m];
    acc0 = __builtin_amdgcn_wmma_f32_16x16x4_f32(false, a, false, b0,
                                                 (short)0, acc0, false, false);
    acc1 = __builtin_amdgcn_wmma_f32_16x16x4_f32(false, a, false, b1v,
                                                 (short)0, acc1, false, false);
  }

  const float bias0 = b1[m], bias1 = b1[m + 16];
#pragma unroll
  for (int r = 0; r < 8; ++r) {
    const int M  = r + hi * 8;
    const int pp = tile * 16 + M;
    out[(size_t)pp * 32 + m]      = fmaxf(acc0[r] + bias0, 0.f);
    out[(size_t)pp * 32 + 16 + m] = fmaxf(acc1[r] + bias1, 0.f);
  }
}

// =====================================================================
// 2x2 stride-2 VALID maxpool, NHWC, one thread per output element
// =====================================================================
__global__ void maxpool_k(const float* __restrict__ in, float* __restrict__ out,
                          int IW, int OW, int C, int total) {
  int idx = blockIdx.x * blockDim.x + threadIdx.x;
  if (idx >= total) return;
  int c = idx % C;
  int t = idx / C;
  int ox = t % OW, oy = t / OW;
  const float* p0 = in + (size_t)((oy * 2) * IW + ox * 2) * C + c;
  float m = fmaxf(fmaxf(p0[0], p0[C]),
                  fmaxf(p0[(size_t)IW * C], p0[(size_t)IW * C + C]));
  out[idx] = m;
}

// =====================================================================
// conv2: (98,148,32) -> (32,49,64), 3x3 stride 3, direct
// =====================================================================
__global__ void conv2_k(const float* __restrict__ in, const float* __restrict__ w,
                        const float* __restrict__ b, float* __restrict__ out) {
  int idx = blockIdx.x * blockDim.x + threadIdx.x;
  if (idx >= C2_OH * C2_OW * 64) return;
  int oc = idx & 63;
  int t  = idx >> 6;
  int ox = t % C2_OW, oy = t / C2_OW;
  float s = b[oc];
  for (int kh = 0; kh < 3; ++kh)
    for (int kw = 0; kw < 3; ++kw) {
      const float* ip = in + (size_t)((oy * 3 + kh) * P1_W + ox * 3 + kw) * 32;
      const float* wp = w + (size_t)((kh * 3 + kw) * 32) * 64 + oc;
#pragma unroll 8
      for (int ic = 0; ic < 32; ++ic) s += ip[ic] * wp[(size_t)ic * 64];
    }
  out[idx] = fmaxf(s, 0.f);
}

// =====================================================================
// conv3: (16,24,64) -> (5,8,128), 3x3 stride 3, direct
// =====================================================================
__global__ void conv3_k(const float* __restrict__ in, const float* __restrict__ w,
                        const float* __restrict__ b, float* __restrict__ out) {
  int idx = blockIdx.x * blockDim.x + threadIdx.x;
  if (idx >= C3_OH * C3_OW * 128) return;
  int oc = idx & 127;
  int t  = idx >> 7;
  int ox = t % C3_OW, oy = t / C3_OW;
  float s = b[oc];
  for (int kh = 0; kh < 3; ++kh)
    for (int kw = 0; kw < 3; ++kw) {
      const float* ip = in + (size_t)((oy * 3 + kh) * P2_W + ox * 3 + kw) * 64;
      const float* wp = w + (size_t)((kh * 3 + kw) * 64) * 128 + oc;
#pragma unroll 8
      for (int ic = 0; ic < 64; ++ic) s += ip[ic] * wp[(size_t)ic * 128];
    }
  out[idx] = fmaxf(s, 0.f);
}

// =====================================================================
// Dense GEMV, split-K partial sums. Streams 786 MB of wd with b128 loads
// (4 consecutive columns per thread -> 512B per wave per issue).
// grid = (38, SPLITK), block = 256. part[ks][n] = sum_k x[k]*wd[k][n].
// x chunk staged into LDS via async global->LDS when available.
// =====================================================================
__global__ __launch_bounds__(256) void gemv_partial(
    const float* __restrict__ x, const float* __restrict__ wd,
    float* __restrict__ part) {
  __shared__ float xs[KCH];
  const int kb = blockIdx.y * KCH;
#if defined(HAVE_ASYNC_LDS)
  for (int t = threadIdx.x; t < KCH; t += 256) {
    gbl_int* g = (gbl_int*)(unsigned long long)(x + kb + t);
    lds_int* l = (lds_int*)(unsigned)(unsigned long long)(void*)&xs[t];
    __builtin_amdgcn_global_load_async_to_lds_b32(g, l, 0, 0);
  }
  __builtin_amdgcn_s_wait_asynccnt(0);
#else
  for (int t = threadIdx.x; t < KCH; t += 256) xs[t] = x[kb + t];
#endif
  __syncthreads();

  const int n4 = blockIdx.x * 256 + threadIdx.x;  // float4 column group
  if (n4 >= DCOLS4) return;
  float4 acc = make_float4(0.f, 0.f, 0.f, 0.f);
  const float4* wp = (const float4*)(wd + (size_t)kb * DCOLS) + n4;
#pragma unroll 4
  for (int k = 0; k < KCH; ++k) {
    if ((k & 63) == 0)
      __builtin_prefetch(wp + (size_t)(k + 96) * DCOLS4, 0, 0);
    const float4 w = wp[(size_t)k * DCOLS4];
    const float xv = xs[k];
    acc.x = fmaf(xv, w.x, acc.x);
    acc.y = fmaf(xv, w.y, acc.y);
    acc.z = fmaf(xv, w.z, acc.z);
    acc.w = fmaf(xv, w.w, acc.w);
  }
  ((float4*)(part + (size_t)blockIdx.y * DCOLS))[n4] = acc;
}

__global__ void gemv_reduce(const float* __restrict__ part,
                            const float* __restrict__ bd,
                            float* __restrict__ xout) {
  int n4 = blockIdx.x * blockDim.x + threadIdx.x;
  if (n4 >= DCOLS4) return;
  float4 s = ((const float4*)bd)[n4];
#pragma unroll
  for (int p = 0; p < SPLITK; ++p) {
    float4 v = ((const float4*)(part + (size_t)p * DCOLS))[n4];
    s.x += v.x; s.y += v.y; s.z += v.z; s.w += v.w;
  }
  float4 r;
  r.x = fmaxf(s.x, 0.f); r.y = fmaxf(s.y, 0.f);
  r.z = fmaxf(s.z, 0.f); r.w = fmaxf(s.w, 0.f);
  ((float4*)xout)[n4] = r;
}

// =====================================================================
// filter_boxes front half: build ip = [x,y,x2,y2,conf,argmax,maxscore]
// =====================================================================
__global__ void filter_k(const float* __restrict__ x, float* __restrict__ ipb) {
  int i = blockIdx.x * blockDim.x + threadIdx.x;
  if (i >= NB) return;
  float p[VL];
#pragma unroll
  for (int j = 0; j < VL; ++j) p[j] = x[(size_t)i * VL + j];
  float mask = (p[4] > 0.5f) ? 1.f : 0.f;
#pragma unroll
  for (int j = 0; j < VL; ++j) p[j] *= mask;
  float ms = p[5];
  int mi = 0;
  if (p[6] > ms) { ms = p[6]; mi = 1; }
  if (p[7] > ms) { ms = p[7]; mi = 2; }
#pragma unroll
  for (int j = 0; j < 5; ++j) ipb[(size_t)i * 7 + j] = p[j];
  ipb[(size_t)i * 7 + 5] = (float)mi;
  ipb[(size_t)i * 7 + 6] = ms;
}

// =====================================================================
// Per-class greedy NMS. One 1024-thread block per class.
// Bitonic sort of packed keys (descending score, ascending index) exactly
// reproduces stable argsort(-scores); pads (idx>=4800) sort after all
// real entries. Then the sequential suppression scan.
// =====================================================================
__device__ __forceinline__ unsigned int fordkey(float f) {
  unsigned u = __float_as_uint(f);
  return (u & 0x80000000u) ? ~u : (u | 0x80000000u);  // ascending order map
}

__global__ __launch_bounds__(1024) void nms_class(
    const float* __restrict__ ipb, int* __restrict__ orderbuf,
    float* __restrict__ boxesbuf, int* __restrict__ keepc) {
  const int c = blockIdx.x;
  __shared__ unsigned long long keys[SORTN];
  __shared__ unsigned int alive[NB];

  const unsigned int kinf = ~fordkey(-__builtin_inff());

  for (int t = threadIdx.x; t < SORTN; t += 1024) {
    float score = -__builtin_inff();
    if (t < NB) {
      float p0 = ipb[(size_t)t * 7 + 0];
      float p5 = ipb[(size_t)t * 7 + 5];
      float p6 = ipb[(size_t)t * 7 + 6];
      if ((p0 != 0.0f) && (p5 == (float)c) && (p6 != 0.0f))
        score = ipb[(size_t)t * 7 + 4];
    }
    unsigned int sk = ~fordkey(score);  // descending score
    keys[t] = ((unsigned long long)sk << 13) | (unsigned long long)t;
  }

  // bitonic sort ascending
  for (unsigned k = 2; k <= SORTN; k <<= 1) {
    for (unsigned j = k >> 1; j > 0; j >>= 1) {
      __syncthreads();
      for (int i = threadIdx.x; i < SORTN; i += 1024) {
        int ixj = i ^ (int)j;
        if (ixj > i) {
          unsigned long long a = keys[i], b = keys[ixj];
          bool up = ((i & (int)k) == 0);
          if ((a > b) == up) { keys[i] = b; keys[ixj] = a; }
        }
      }
    }
  }
  __syncthreads();

  int* myorder = orderbuf + (size_t)c * NB;
  float* mybox = boxesbuf + (size_t)c * NB * 4;
  for (int sp = threadIdx.x; sp < NB; sp += 1024) {
    int oi = (int)(keys[sp] & 0x1FFFull);
    myorder[sp] = oi;
#pragma unroll
    for (int j = 0; j < 4; ++j) mybox[(size_t)sp * 4 + j] = ipb[(size_t)oi * 7 + j];
    unsigned int sk = (unsigned int)(keys[sp] >> 13);
    alive[sp] = (sk != kinf) ? 1u : 0u;
  }

  // sequential greedy suppression
  for (int i = 0; i < NB - 1; ++i) {
    __syncthreads();
    if (!alive[i]) continue;  // uniform: same LDS value for all threads
    float bx1 = mybox[(size_t)i * 4 + 0], by1 = mybox[(size_t)i * 4 + 1];
    float bx2 = mybox[(size_t)i * 4 + 2], by2 = mybox[(size_t)i * 4 + 3];
    float a1 = (bx2 - bx1 + 1.f) * (by2 - by1 + 1.f);
    for (int j = i + 1 + (int)threadIdx.x; j < NB; j += 1024) {
      float cx1 = mybox[(size_t)j * 4 + 0], cy1 = mybox[(size_t)j * 4 + 1];
      float cx2 = mybox[(size_t)j * 4 + 2], cy2 = mybox[(size_t)j * 4 + 3];
      float ix1 = fmaxf(bx1, cx1), iy1 = fmaxf(by1, cy1);
      float ix2 = fminf(bx2, cx2), iy2 = fminf(by2, cy2);
      float inter = fmaxf(ix2 - ix1 + 1.f, 0.f) * fmaxf(iy2 - iy1 + 1.f, 0.f);
      float a2 = (cx2 - cx1 + 1.f) * (cy2 - cy1 + 1.f);
      float iou = inter / (a1 + a2 - inter + 1e-16f);
      if (iou >= 0.4f) alive[j] = 0u;
    }
  }
  __syncthreads();

  for (int sp = threadIdx.x; sp < NB; sp += 1024)
    keepc[(size_t)c * NB + myorder[sp]] = (int)alive[sp];
}

// =====================================================================
// keep = any over classes; filtered = ip * keep; keep as 0/1 float
// =====================================================================
__global__ void combine_k(const float* __restrict__ ipb,
                          const int* __restrict__ keepc,
                          float* __restrict__ outF, float* __restrict__ outK) {
  int i = blockIdx.x * blockDim.x + threadIdx.x;
  if (i >= NB) return;
  int k = keepc[i] | keepc[NB + i] | keepc[2 * NB + i];
  float kf = k ? 1.f : 0.f;
#pragma unroll
  for (int j = 0; j < 7; ++j) outF[(size_t)i * 7 + j] = ipb[(size_t)i * 7 + j] * kf;
  outK[i] = kf;
}

// =====================================================================
extern "C" void kernel_launch(void* const* d_in, const int* in_sizes, int n_in,
                              void* d_out, int out_size, void* d_ws, size_t ws_size,
                              hipStream_t stream) {
  const float* img = (const float*)d_in[0];
  const float* w1  = (const float*)d_in[1];
  const float* b1  = (const float*)d_in[2];
  const float* w2  = (const float*)d_in[3];
  const float* b2  = (const float*)d_in[4];
  const float* w3  = (const float*)d_in[5];
  const float* b3  = (const float*)d_in[6];
  const float* wd  = (const float*)d_in[7];
  const float* bd  = (const float*)d_in[8];

  float* out_x    = (float*)d_out;             // 38400
  float* out_filt = out_x + DCOLS;             // 33600
  float* out_keep = out_filt + NB * 7;         // 4800

  // workspace layout (floats)
  float* ws = (float*)d_ws;
  size_t o = 0;
  float* conv1o = ws + o; o += (size_t)C1_PIX * 32;         // 1,856,512
  float* pool1o = ws + o; o += (size_t)P1_H * P1_W * 32;    //   464,128
  float* conv2o = ws + o; o += (size_t)C2_OH * C2_OW * 64;  //   100,352
  float* pool2o = ws + o; o += (size_t)P2_H * P2_W * 64;    //    24,576
  float* conv3o = ws + o; o += DK;                          //     5,120
  float* partb  = ws + o; o += (size_t)SPLITK * DCOLS;      //   614,400
  float* ipb    = ws + o; o += (size_t)NB * 7;              //    33,600
  float* boxesb = ws + o; o += (size_t)3 * NB * 4;          //    57,600
  int*   orderb = (int*)(ws + o); o += (size_t)3 * NB;      //    14,400
  int*   keepcb = (int*)(ws + o); o += (size_t)3 * NB;      //    14,400

  // conv1 (WMMA) + relu
  conv1_wmma<<<(C1_TILES + 7) / 8, 256, 0, stream>>>(img, w1, b1, conv1o);
  // maxpool 1
  {
    int total = P1_H * P1_W * 32;
    maxpool_k<<<(total + 255) / 256, 256, 0, stream>>>(conv1o, pool1o, C1_OW,
                                                       P1_W, 32, total);
  }
  // conv2 + relu
  conv2_k<<<(C2_OH * C2_OW * 64 + 255) / 256, 256, 0, stream>>>(pool1o, w2, b2,
                                                                conv2o);
  // maxpool 2
  {
    int total = P2_H * P2_W * 64;
    maxpool_k<<<(total + 255) / 256, 256, 0, stream>>>(conv2o, pool2o, C2_OW,
                                                       P2_W, 64, total);
  }
  // conv3 + relu
  conv3_k<<<(C3_OH * C3_OW * 128 + 255) / 256, 256, 0, stream>>>(pool2o, w3, b3,
                                                                 conv3o);
  // dense GEMV (bandwidth dominant): split-K partials, then reduce+bias+relu
  gemv_partial<<<dim3((DCOLS4 + 255) / 256, SPLITK), 256, 0, stream>>>(conv3o,
                                                                       wd, partb);
  gemv_reduce<<<(DCOLS4 + 255) / 256, 256, 0, stream>>>(partb, bd, out_x);
  // box filtering
  filter_k<<<(NB + 255) / 256, 256, 0, stream>>>(out_x, ipb);
  // per-class NMS (3 classes)
  nms_class<<<3, 1024, 0, stream>>>(ipb, orderb, boxesb, keepcb);
  // combine keep + write filtered outputs
  combine_k<<<(NB + 255) / 256, 256, 0, stream>>>(ipb, keepcb, out_filt, out_keep);
}